// re_unit_54022098649172
// MI455X (gfx1250) — compile-verified
//
#include <hip/hip_runtime.h>
#include <hip/hip_bf16.h>
#include <math.h>

typedef __attribute__((ext_vector_type(2))) float v2f;
typedef __attribute__((ext_vector_type(8))) float v8f;
typedef __attribute__((ext_vector_type(2))) int   i32x2;
typedef __attribute__((ext_vector_type(4))) int   i32x4;

#define LL 128
#define BB 4
#define HH 768
#define RR 97
#define LN_EPS 1e-5f
#define XSTR 770   // LN-tile row stride (dwords): banks (2m+k)%64 distinct across lanes
#define GSTR 68    // gemm X-tile row stride: 16B-aligned rows, banks (4r+k)%64 distinct

// ---------- CDNA5 async global->LDS staging (guarded; falls back to sync copies) ----------
#if defined(__gfx1250__) && __has_builtin(__builtin_amdgcn_global_load_async_to_lds_b64) && \
    __has_builtin(__builtin_amdgcn_global_load_async_to_lds_b128)
#define USE_ASYNC 1

// builtin prototypes (probe-confirmed): (v2i/v4i addrspace(1)*, v2i/v4i addrspace(3)*, imm, imm)
static __device__ __forceinline__ void async_cp_b64(const float* g, float* l) {
  __builtin_amdgcn_global_load_async_to_lds_b64(
      (__attribute__((address_space(1))) i32x2*)(uint64_t)(uintptr_t)g,
      (__attribute__((address_space(3))) i32x2*)(uint32_t)(uintptr_t)l, 0, 0);
}
static __device__ __forceinline__ void async_cp_b128(const float* g, float* l) {
  __builtin_amdgcn_global_load_async_to_lds_b128(
      (__attribute__((address_space(1))) i32x4*)(uint64_t)(uintptr_t)g,
      (__attribute__((address_space(3))) i32x4*)(uint32_t)(uintptr_t)l, 0, 0);
}
static __device__ __forceinline__ void wait_async0() {
#if __has_builtin(__builtin_amdgcn_s_wait_asynccnt)
  __builtin_amdgcn_s_wait_asynccnt(0);
#else
  asm volatile("s_wait_asynccnt 0x0" ::: "memory");
#endif
}
#else
#define USE_ASYNC 0
#endif

// ---------------- concat h1||h2 into row-major (512 x 1536) ----------------
__global__ void concat_kernel(const float* __restrict__ h1, const float* __restrict__ h2,
                              float* __restrict__ xcat, int total) {
  int i = blockIdx.x * blockDim.x + threadIdx.x;
  if (i >= total) return;
  int row = i / (2 * HH);
  int c   = i - row * (2 * HH);
  xcat[i] = (c < HH) ? h1[(size_t)row * HH + c] : h2[(size_t)row * HH + (c - HH)];
}

// ------- generic WMMA f32 GEMM: C[M,N] = act(X[M,K] @ W[N,K]^T + bias[n]) -------
// W rows have stride ldw (supports Wh sub-matrices). grid=(M/64, N/16), block=128 (4 waves).
// W tile stored K-pair-interleaved: wsm[(k>>1)*32 + 2n + (k&1)] -> B fragment = one ds_load_b64.
__global__ __launch_bounds__(128)
void gemm_xtw(const float* __restrict__ X, const float* __restrict__ W,
              const float* __restrict__ bias, float* __restrict__ C,
              int M, int N, int K, int ldw, int act) {
  __shared__ float xs[64 * GSTR];  // 64 rows x 64-k chunk
  __shared__ float wsm[32 * 32];   // 64 k x 16 n, pair-interleaved
  const int t    = threadIdx.x;
  const int wave = t >> 5;
  const int lane = t & 31;
  const int half = lane >> 4;
  const int l16  = lane & 15;
  const int m0   = blockIdx.x * 64;
  const int n0   = blockIdx.y * 16;

  v8f acc = {};
  for (int kc = 0; kc < K; kc += 64) {
#if USE_ASYNC
    // X tile: 64 rows x 16 quads, contiguous 16B on both sides
    for (int j = t; j < 64 * 16; j += 128) {
      int r = j >> 4, q = j & 15;
      async_cp_b128(&X[(size_t)(m0 + r) * K + kc + 4 * q], &xs[r * GSTR + 4 * q]);
    }
    // W tile: 16 n x 32 k-pairs
    for (int j = t; j < 16 * 32; j += 128) {
      int n = j >> 5, p = j & 31;
      async_cp_b64(&W[(size_t)(n0 + n) * ldw + kc + 2 * p], &wsm[p * 32 + 2 * n]);
    }
    wait_async0();
#else
    for (int j = t; j < 64 * 64; j += 128) {
      int r = j >> 6, kk = j & 63;
      xs[r * GSTR + kk] = X[(size_t)(m0 + r) * K + kc + kk];
    }
    for (int j = t; j < 64 * 16; j += 128) {
      int n = j >> 6, kk = j & 63;
      wsm[(kk >> 1) * 32 + 2 * n + (kk & 1)] = W[(size_t)(n0 + n) * ldw + kc + kk];
    }
#endif
    __syncthreads();
    const float* xrow = &xs[(wave * 16 + l16) * GSTR];
    for (int kk = 0; kk < 64; kk += 4) {
      int kb = kk + 2 * half;          // lanes 0-15: K=kk,kk+1 ; lanes 16-31: K=kk+2,kk+3
      int kp = kb >> 1;
      v2f af, bf;
      af.x = xrow[kb];
      af.y = xrow[kb + 1];
      bf.x = wsm[kp * 32 + 2 * l16];
      bf.y = wsm[kp * 32 + 2 * l16 + 1];
      acc = __builtin_amdgcn_wmma_f32_16x16x4_f32(false, af, false, bf, (short)0, acc, false, false);
    }
    __syncthreads();
  }
  int ng = n0 + l16;
  float bval = bias ? bias[ng] : 0.0f;
  for (int v = 0; v < 8; v++) {
    int mg = m0 + wave * 16 + v + 8 * half;  // D layout: vgpr v -> M=v (lanes<16) / v+8
    float val = acc[v] + bval;
    if (act == 1) val = tanhf(val);
    C[(size_t)mg * N + ng] = val;
  }
}

// ---------------- max over L:  g[b,h] = max_l gpre[l,b,h] ----------------
__global__ void maxred_kernel(const float* __restrict__ gpre, float* __restrict__ g) {
  int i = blockIdx.x * blockDim.x + threadIdx.x;
  if (i >= BB * HH) return;
  int b = i / HH, h = i - b * HH;
  float m = -3.4e38f;
  for (int l = 0; l < LL; l++) m = fmaxf(m, gpre[((size_t)l * BB + b) * HH + h]);
  g[i] = m;
}

// ---------------- gp[b,h] = g[b,:] . Wh[h, 2H:3H] + bh[h] ----------------
__global__ void gp_kernel(const float* __restrict__ g, const float* __restrict__ Wh,
                          const float* __restrict__ bh, float* __restrict__ gp) {
  int i = blockIdx.x * blockDim.x + threadIdx.x;
  if (i >= BB * HH) return;
  int b = i / HH, h = i - b * HH;
  const float* grow = &g[(size_t)b * HH];
  const float* wrow = &Wh[(size_t)h * (3 * HH) + 2 * HH];
  float s = 0.f;
  for (int k = 0; k < HH; k++) s += grow[k] * wrow[k];
  gp[i] = s + bh[h];
}

// ------- fused: pre=a+b2+gp -> LayerNorm -> ELU -> x@Wo^T + bo -> sigmoid -> mask -------
// 1024 blocks x 64 rows; block=128 threads (4 waves, 16 rows per wave).
__global__ __launch_bounds__(128)
void fused_kernel(const float* __restrict__ A, const float* __restrict__ B2,
                  const float* __restrict__ GP,
                  const float* __restrict__ gamma, const float* __restrict__ beta,
                  const float* __restrict__ Wo, const float* __restrict__ bo,
                  const float* __restrict__ mask, float* __restrict__ out) {
  extern __shared__ float sm[];
  float* xt = sm;                 // 64 * XSTR  (LN'd activations)
  float* bt = sm + 64 * XSTR;     // 384 k-pairs * (16 n * 2)  (pair-interleaved Wo tile)
  const int t    = threadIdx.x;
  const int wave = t >> 5;
  const int lane = t & 31;
  const int half = lane >> 4;
  const int l16  = lane & 15;
  const int row0 = blockIdx.x * 64;

  // ---- stage 1: LayerNorm + ELU, 16 rows per wave ----
  for (int i = 0; i < 16; i++) {
    int rl  = wave * 16 + i;
    int rg  = row0 + rl;            // rg = (k*128 + l)*4 + b
    int k   = rg >> 9;
    int rem = rg & 511;
    int l   = rem >> 2;
    int b   = rem & 3;
    const float* arow = &A[((size_t)k * BB + b) * HH];
    const float* brow = &B2[((size_t)l * BB + b) * HH];
    const float* grow = &GP[(size_t)b * HH];
    float s = 0.f, s2 = 0.f;
    for (int h = lane; h < HH; h += 32) {
      float p = arow[h] + brow[h] + grow[h];
      xt[rl * XSTR + h] = p;
      s += p; s2 += p * p;
    }
    for (int o = 16; o > 0; o >>= 1) {   // wave32 tree reduction
      s  += __shfl_xor(s,  o, 32);
      s2 += __shfl_xor(s2, o, 32);
    }
    float mean = s * (1.0f / HH);
    float var  = s2 * (1.0f / HH) - mean * mean;
    float rstd = rsqrtf(var + LN_EPS);
    for (int h = lane; h < HH; h += 32) {
      float x = (xt[rl * XSTR + h] - mean) * rstd * gamma[h] + beta[h];
      x = (x > 0.f) ? x : (expf(x) - 1.f);   // ELU
      xt[rl * XSTR + h] = x;
    }
  }
  __syncthreads();

  // ---- stage 2: x(64x768) @ Wo^T via WMMA, 7 N-tiles of 16 (R=97 padded to 112) ----
  for (int nt = 0; nt < 7; nt++) {
#if USE_ASYNC
    for (int j = t; j < 16 * 384; j += 128) {   // 16 n x 384 k-pairs
      int n = j / 384;
      int p = j - n * 384;
      int r = nt * 16 + n;
      if (r < RR) {
        async_cp_b64(&Wo[(size_t)r * HH + 2 * p], &bt[p * 32 + 2 * n]);
      } else {
        bt[p * 32 + 2 * n]     = 0.f;           // DMA can't synthesize zeros
        bt[p * 32 + 2 * n + 1] = 0.f;
      }
    }
    wait_async0();
#else
    for (int j = t; j < HH * 16; j += 128) {
      int n  = j / HH;
      int kk = j - n * HH;
      int r  = nt * 16 + n;
      bt[(kk >> 1) * 32 + 2 * n + (kk & 1)] = (r < RR) ? Wo[(size_t)r * HH + kk] : 0.f;
    }
#endif
    __syncthreads();
    v8f acc = {};
    const float* xrow = &xt[(wave * 16 + l16) * XSTR];
    for (int kk = 0; kk < HH; kk += 4) {
      int kb = kk + 2 * half;
      int kp = kb >> 1;
      v2f af, bf;
      af.x = xrow[kb];
      af.y = xrow[kb + 1];
      bf.x = bt[kp * 32 + 2 * l16];
      bf.y = bt[kp * 32 + 2 * l16 + 1];
      acc = __builtin_amdgcn_wmma_f32_16x16x4_f32(false, af, false, bf, (short)0, acc, false, false);
    }
    int r = nt * 16 + l16;
    float bov = (r < RR) ? bo[r] : 0.f;
    for (int v = 0; v < 8; v++) {
      int rl  = wave * 16 + v + 8 * half;
      int rg  = row0 + rl;
      int k   = rg >> 9;
      int rem = rg & 511;
      int l   = rem >> 2;
      int b   = rem & 3;
      if (r < RR) {
        float val = 1.f / (1.f + expf(-(acc[v] + bov)));
        float m   = mask[k * BB + b] * mask[l * BB + b];
        out[(size_t)rg * RR + r] = val * m;
      }
    }
    __syncthreads();
  }
}

extern "C" void kernel_launch(void* const* d_in, const int* in_sizes, int n_in,
                              void* d_out, int out_size, void* d_ws, size_t ws_size,
                              hipStream_t stream) {
  const float* h1    = (const float*)d_in[0];
  const float* h2    = (const float*)d_in[1];
  const float* mask  = (const float*)d_in[2];
  const float* Wr    = (const float*)d_in[3];
  const float* br    = (const float*)d_in[4];
  const float* Wh    = (const float*)d_in[5];
  const float* bh    = (const float*)d_in[6];
  const float* gamma = (const float*)d_in[7];
  const float* beta  = (const float*)d_in[8];
  const float* Wo    = (const float*)d_in[9];
  const float* bo    = (const float*)d_in[10];
  float* out = (float*)d_out;

  float* ws    = (float*)d_ws;
  float* xcat  = ws;                       // 512*1536
  float* gpre  = xcat + 512 * 2 * HH;      // 512*768
  float* aarr  = gpre + 512 * HH;          // 512*768
  float* b2arr = aarr + 512 * HH;          // 512*768
  float* g     = b2arr + 512 * HH;         // 4*768
  float* gp    = g + BB * HH;              // 4*768

  int totcat = 512 * 2 * HH;
  concat_kernel<<<(totcat + 255) / 256, 256, 0, stream>>>(h1, h2, xcat, totcat);

  // g_pre = tanh(Xcat @ Wr^T + br) : M=512, N=768, K=1536
  gemm_xtw<<<dim3(8, 48), 128, 0, stream>>>(xcat, Wr, br, gpre, 512, HH, 2 * HH, 2 * HH, 1);
  // a  = h2 @ W1^T : Wh[:, 0:768], row stride 2304
  gemm_xtw<<<dim3(8, 48), 128, 0, stream>>>(h2, Wh, nullptr, aarr, 512, HH, HH, 3 * HH, 0);
  // b2 = h2 @ W2^T : Wh[:, 768:1536]
  gemm_xtw<<<dim3(8, 48), 128, 0, stream>>>(h2, Wh + HH, nullptr, b2arr, 512, HH, HH, 3 * HH, 0);

  maxred_kernel<<<(BB * HH + 255) / 256, 256, 0, stream>>>(gpre, g);
  gp_kernel<<<(BB * HH + 255) / 256, 256, 0, stream>>>(g, Wh, bh, gp);

  size_t smem = (size_t)(64 * XSTR + (HH / 2) * 32) * sizeof(float);  // ~240.5 KB of 320 KB WGP LDS
  fused_kernel<<<(LL * LL * BB) / 64, 128, smem, stream>>>(aarr, b2arr, gp, gamma, beta,
                                                           Wo, bo, mask, out);
}